// LowFrequencyEncoder_79903571574980
// MI455X (gfx1250) — compile-verified
//
#include <hip/hip_runtime.h>

typedef __attribute__((ext_vector_type(2))) float v2f;
typedef __attribute__((ext_vector_type(8))) float v8f;

#define GCN_D 64
#define BN_EPS 1e-5f

// -------------------- degree / normalization --------------------
__global__ void deg_init_kernel(unsigned int* __restrict__ deg, int N) {
  int i = blockIdx.x * blockDim.x + threadIdx.x;
  if (i < N) deg[i] = 1u;  // self loop
}

__global__ void deg_count_kernel(const long long* __restrict__ dst,
                                 unsigned int* __restrict__ deg, int E) {
  int e = blockIdx.x * blockDim.x + threadIdx.x;
  if (e < E) atomicAdd(&deg[(int)dst[e]], 1u);
}

__global__ void dinv_kernel(const unsigned int* __restrict__ deg,
                            float* __restrict__ dinv, int N) {
  int i = blockIdx.x * blockDim.x + threadIdx.x;
  if (i < N) dinv[i] = rsqrtf((float)deg[i]);
}

// -------------------- GEMM: HW = H @ W  (fp32 WMMA 16x16x4) --------------------
// Block = 256 threads = 8 waves; each wave computes a 16x64 row block.
// W staged in LDS K-pair-packed: Wp[p*64+n] = {W[2p][n], W[2p+1][n]} so each
// B fragment is one aligned ds_load_b64 directly into an even VGPR pair.
__global__ __launch_bounds__(256) void gemm64_wmma(const float* __restrict__ H,
                                                   const float* __restrict__ W,
                                                   float* __restrict__ HW, int N) {
  __shared__ v2f Wp[32 * GCN_D];  // 16 KB
  for (int t = threadIdx.x; t < 32 * GCN_D; t += 256) {
    int p = t >> 6, n = t & 63;
    v2f w2;
    w2.x = W[p * 128 + n];        // W[2p][n]
    w2.y = W[p * 128 + 64 + n];   // W[2p+1][n]
    Wp[t] = w2;
  }
  __syncthreads();

  const int wave = threadIdx.x >> 5;   // 0..7
  const int lane = threadIdx.x & 31;
  const int half = lane >> 4;          // 0: lanes 0-15, 1: lanes 16-31
  const int l    = lane & 15;
  const int row0 = blockIdx.x * 128 + wave * 16;
  if (row0 + 16 > N) return;           // wave-uniform (N % 16 == 0)

  // A fragments: 16x4 f32 per step. ISA layout:
  //   lanes 0-15:  M=lane,    v0 -> K=0, v1 -> K=1
  //   lanes16-31:  M=lane-16, v0 -> K=2, v1 -> K=3
  v2f a[16];
  const float* hrow = H + (size_t)(row0 + l) * GCN_D + 2 * half;
#pragma unroll
  for (int s = 0; s < 16; ++s) {
    a[s].x = hrow[s * 4 + 0];
    a[s].y = hrow[s * 4 + 1];
  }

#pragma unroll
  for (int cb = 0; cb < 4; ++cb) {
    const int n0 = cb * 16;
    v8f acc = {0.f, 0.f, 0.f, 0.f, 0.f, 0.f, 0.f, 0.f};
#pragma unroll
    for (int s = 0; s < 16; ++s) {
      // B fragment: v0 -> K = 4s+2*half, v1 -> K+1  == packed pair (2s+half)
      v2f b = Wp[(s * 2 + half) * GCN_D + n0 + l];
      acc = __builtin_amdgcn_wmma_f32_16x16x4_f32(false, a[s], false, b,
                                                  (short)0, acc, false, false);
    }
    // C/D layout: VGPR r -> M = r + 8*half, N = n0 + l
    float* orow = HW + (size_t)(row0 + half * 8) * GCN_D + n0 + l;
#pragma unroll
    for (int r = 0; r < 8; ++r) orow[(size_t)r * GCN_D] = acc[r];
  }
}

// -------------------- agg init: bias + self-loop term --------------------
// agg[i][f] = bias[f] + hW[i][f] * dinv[i]^2   (self loop has norm = dinv^2)
__global__ void selfloop_bias_kernel(const float* __restrict__ hW,
                                     const float* __restrict__ dinv,
                                     const float* __restrict__ bias,
                                     float* __restrict__ agg, int NtimesD) {
  int t = blockIdx.x * blockDim.x + threadIdx.x;
  if (t < NtimesD) {
    int i = t >> 6, f = t & 63;
    float di = dinv[i];
    agg[t] = bias[f] + hW[t] * di * di;
  }
}

// -------------------- edge scatter: agg[dst] += hW[src] * norm --------------------
// 16 consecutive threads share one edge; each handles 4 features (float4 gather).
__global__ void scatter_kernel(const float* __restrict__ hW,
                               const long long* __restrict__ src,
                               const long long* __restrict__ dst,
                               const float* __restrict__ dinv,
                               float* __restrict__ agg, int E) {
  int t = blockIdx.x * blockDim.x + threadIdx.x;
  int e = t >> 4, c = t & 15;
  if (e >= E) return;
  int s = (int)src[e];
  int d = (int)dst[e];
  float w = dinv[s] * dinv[d];
  const float4 v = *(const float4*)(hW + (size_t)s * GCN_D + c * 4);
  float* p = agg + (size_t)d * GCN_D + c * 4;
  __hip_atomic_fetch_add(p + 0, v.x * w, __ATOMIC_RELAXED, __HIP_MEMORY_SCOPE_AGENT);
  __hip_atomic_fetch_add(p + 1, v.y * w, __ATOMIC_RELAXED, __HIP_MEMORY_SCOPE_AGENT);
  __hip_atomic_fetch_add(p + 2, v.z * w, __ATOMIC_RELAXED, __HIP_MEMORY_SCOPE_AGENT);
  __hip_atomic_fetch_add(p + 3, v.w * w, __ATOMIC_RELAXED, __HIP_MEMORY_SCOPE_AGENT);
}

// -------------------- fused BatchNorm (eval) + ReLU --------------------
__global__ void bn_relu_kernel(const float* __restrict__ agg,
                               const float* __restrict__ g,
                               const float* __restrict__ be,
                               const float* __restrict__ m,
                               const float* __restrict__ v,
                               float* __restrict__ out, int NtimesD) {
  int t = blockIdx.x * blockDim.x + threadIdx.x;
  if (t < NtimesD) {
    int f = t & 63;
    float y = (agg[t] - m[f]) * rsqrtf(v[f] + BN_EPS) * g[f] + be[f];
    out[t] = y > 0.f ? y : 0.f;
  }
}

extern "C" void kernel_launch(void* const* d_in, const int* in_sizes, int n_in,
                              void* d_out, int out_size, void* d_ws, size_t ws_size,
                              hipStream_t stream) {
  const float*     x  = (const float*)d_in[0];
  const long long* ei = (const long long*)d_in[1];
  const float* W1 = (const float*)d_in[2];
  const float* b1 = (const float*)d_in[3];
  const float* g1 = (const float*)d_in[4];
  const float* be1 = (const float*)d_in[5];
  const float* m1 = (const float*)d_in[6];
  const float* v1 = (const float*)d_in[7];
  const float* W2 = (const float*)d_in[8];
  const float* b2 = (const float*)d_in[9];
  const float* g2 = (const float*)d_in[10];
  const float* be2 = (const float*)d_in[11];
  const float* m2 = (const float*)d_in[12];
  const float* v2 = (const float*)d_in[13];
  const float* W3 = (const float*)d_in[14];
  const float* b3 = (const float*)d_in[15];

  const int N = in_sizes[0] / GCN_D;  // 50000
  const int E = in_sizes[1] / 2;      // 800000
  const long long* srcIdx = ei;       // edge_index[0]
  const long long* dstIdx = ei + E;   // edge_index[1]

  // workspace layout
  char* ws = (char*)d_ws;
  size_t off = 0;
  auto take = [&](size_t bytes) -> void* {
    void* p = ws + off;
    off += (bytes + 255) & ~(size_t)255;
    return p;
  };
  unsigned int* deg = (unsigned int*)take((size_t)N * 4);
  float* dinv = (float*)take((size_t)N * 4);
  float* h    = (float*)take((size_t)N * GCN_D * 4);
  float* hW   = (float*)take((size_t)N * GCN_D * 4);
  float* agg  = (float*)take((size_t)N * GCN_D * 4);
  float* out  = (float*)d_out;

  const int ND = N * GCN_D;
  dim3 blk(256);
  const int nb_nodes = (N + 255) / 256;
  const int nb_edges = (E + 255) / 256;
  const int nb_elem  = (ND + 255) / 256;
  const int nb_scat  = (E * 16 + 255) / 256;
  const int nb_gemm  = (N + 127) / 128;

  // ---- normalization coefficients ----
  deg_init_kernel<<<nb_nodes, blk, 0, stream>>>(deg, N);
  deg_count_kernel<<<nb_edges, blk, 0, stream>>>(dstIdx, deg, E);
  dinv_kernel<<<nb_nodes, blk, 0, stream>>>(deg, dinv, N);

  // ---- layer 1 ----
  gemm64_wmma<<<nb_gemm, blk, 0, stream>>>(x, W1, hW, N);
  selfloop_bias_kernel<<<nb_elem, blk, 0, stream>>>(hW, dinv, b1, agg, ND);
  scatter_kernel<<<nb_scat, blk, 0, stream>>>(hW, srcIdx, dstIdx, dinv, agg, E);
  bn_relu_kernel<<<nb_elem, blk, 0, stream>>>(agg, g1, be1, m1, v1, h, ND);

  // ---- layer 2 ----
  gemm64_wmma<<<nb_gemm, blk, 0, stream>>>(h, W2, hW, N);
  selfloop_bias_kernel<<<nb_elem, blk, 0, stream>>>(hW, dinv, b2, agg, ND);
  scatter_kernel<<<nb_scat, blk, 0, stream>>>(hW, srcIdx, dstIdx, dinv, agg, E);
  bn_relu_kernel<<<nb_elem, blk, 0, stream>>>(agg, g2, be2, m2, v2, h, ND);

  // ---- layer 3 (no BN/ReLU; write directly into d_out) ----
  gemm64_wmma<<<nb_gemm, blk, 0, stream>>>(h, W3, hW, N);
  selfloop_bias_kernel<<<nb_elem, blk, 0, stream>>>(hW, dinv, b3, out, ND);
  scatter_kernel<<<nb_scat, blk, 0, stream>>>(hW, srcIdx, dstIdx, dinv, out, E);
}